// RandomMap_27668179321051
// MI455X (gfx1250) — compile-verified
//
#include <hip/hip_runtime.h>
#include <hip/hip_bf16.h>

#define N 4096
#define SINKHORN_ITERS 50
#define NCHUNK 16
#define ROWS_PER_CHUNK (N / NCHUNK) /* 256 */

typedef float v2f __attribute__((ext_vector_type(2)));
typedef float v8f __attribute__((ext_vector_type(8)));

// ---------------------------------------------------------------------------
// C[j] = 0
// ---------------------------------------------------------------------------
__global__ void sinkhorn_init_c(float* __restrict__ C) {
    int j = blockIdx.x * blockDim.x + threadIdx.x;
    if (j < N) C[j] = 0.0f;
}

// ---------------------------------------------------------------------------
// Row pass: R[i] = log( sum_j exp(M[i,j] - C[j]) )
// One block per row; float4 (b128) coalesced row reads; LDS tree reduction.
// ---------------------------------------------------------------------------
__global__ __launch_bounds__(256) void sinkhorn_row_lse(
    const float* __restrict__ M, const float* __restrict__ C,
    float* __restrict__ R) {
    __shared__ float sdata[256];
    const int i = blockIdx.x;
    const int tid = threadIdx.x;
    const float* __restrict__ row = M + (size_t)i * N;

    float acc = 0.0f;
#pragma unroll
    for (int j4 = tid * 4; j4 < N; j4 += 256 * 4) {
        const float4 m = *(const float4*)(row + j4);
        const float4 c = *(const float4*)(C + j4);
        acc += __expf(m.x - c.x);
        acc += __expf(m.y - c.y);
        acc += __expf(m.z - c.z);
        acc += __expf(m.w - c.w);
    }
    sdata[tid] = acc;
    __syncthreads();
    for (int s = 128; s > 0; s >>= 1) {
        if (tid < s) sdata[tid] += sdata[tid + s];
        __syncthreads();
    }
    if (tid == 0) R[i] = __logf(sdata[0]);
}

// ---------------------------------------------------------------------------
// Column pass (partial): partial[chunk][j] = sum_{i in chunk} exp(M[i,j]-R[i])
//
// WMMA-as-reduction: A = ones(16x4) so D[m][n] += sum_k B[k][n].
// Each wave owns 16 columns; B tile = exp values of a 4(row)x16(col) tile.
// Since A is all-ones, the K-lane placement inside B is irrelevant — only the
// N (column) position matters, making this robust to the exact B VGPR layout.
// Column sums are replicated across all rows of D, so d[0] in lanes 0..15 is
// the per-column result: no LDS / cross-lane reduction needed at all.
// Block = 128 threads = 4 waves = 64 columns; grid.y = 16 row chunks.
// ---------------------------------------------------------------------------
__global__ __launch_bounds__(128) void sinkhorn_col_partial(
    const float* __restrict__ M, const float* __restrict__ R,
    float* __restrict__ partial) {
    const int lane = threadIdx.x & 31;
    const int wave = threadIdx.x >> 5;
    const int j0 = blockIdx.x * 64 + wave * 16;
    const int rowBase = blockIdx.y * ROWS_PER_CHUNK;
    const int half = lane >> 4;   // 0 -> rows k,k+1 ; 1 -> rows k+2,k+3
    const int lcol = lane & 15;
    const int col = j0 + lcol;

    v2f a;
    a.x = 1.0f;
    a.y = 1.0f;          // ones A-matrix: WMMA becomes a column-sum engine
    v8f d = {};          // f32 accumulator (column sums, replicated over M)

    const float* __restrict__ mc = M + col;
    for (int k = 0; k < ROWS_PER_CHUNK; k += 4) {
        const int r0 = rowBase + k + 2 * half;
        __builtin_prefetch(&mc[(size_t)(r0 + 8) * N], 0, 3);  // near-scope prefetch
        v2f b;
        b.x = __expf(mc[(size_t)r0 * N] - R[r0]);
        b.y = __expf(mc[(size_t)(r0 + 1) * N] - R[r0 + 1]);
        // (neg_a, A, neg_b, B, c_mod, C, reuse_a, reuse_b)
        d = __builtin_amdgcn_wmma_f32_16x16x4_f32(false, a, false, b,
                                                  (short)0, d, false, false);
    }

    if (half == 0) {
        partial[(size_t)blockIdx.y * N + col] = d[0];
    }
}

// ---------------------------------------------------------------------------
// Column pass (finalize): C[j] = log( sum_chunk partial[chunk][j] )
// ---------------------------------------------------------------------------
__global__ void sinkhorn_col_finalize(const float* __restrict__ partial,
                                      float* __restrict__ C) {
    int j = blockIdx.x * blockDim.x + threadIdx.x;
    if (j >= N) return;
    float s = 0.0f;
#pragma unroll
    for (int c = 0; c < NCHUNK; ++c) s += partial[(size_t)c * N + j];
    C[j] = __logf(s);
}

// ---------------------------------------------------------------------------
// Output: out[i,j] = max(exp(M[i,j] - R[i] - C[j]), 1e-9), vectorized float4
// ---------------------------------------------------------------------------
__global__ __launch_bounds__(256) void sinkhorn_final(
    const float* __restrict__ M, const float* __restrict__ R,
    const float* __restrict__ C, float* __restrict__ out) {
    const int idx = blockIdx.x * blockDim.x + threadIdx.x;  // one float4 each
    const int i = idx >> (12 - 2);                          // idx / (N/4)
    const int j4 = (idx & (N / 4 - 1)) << 2;
    const size_t off = (size_t)i * N + j4;

    const float4 m = *(const float4*)(M + off);
    const float4 c = *(const float4*)(C + j4);
    const float r = R[i];
    float4 o;
    o.x = fmaxf(__expf(m.x - r - c.x), 1e-9f);
    o.y = fmaxf(__expf(m.y - r - c.y), 1e-9f);
    o.z = fmaxf(__expf(m.z - r - c.z), 1e-9f);
    o.w = fmaxf(__expf(m.w - r - c.w), 1e-9f);
    *(float4*)(out + off) = o;
}

// ---------------------------------------------------------------------------
extern "C" void kernel_launch(void* const* d_in, const int* in_sizes, int n_in,
                              void* d_out, int out_size, void* d_ws,
                              size_t ws_size, hipStream_t stream) {
    (void)in_sizes; (void)n_in; (void)out_size; (void)ws_size;

    const float* M = (const float*)d_in[0];
    float* out = (float*)d_out;

    char* ws = (char*)d_ws;
    float* R = (float*)(ws);                   // 4096 floats
    float* C = (float*)(ws + 16 * 1024);       // 4096 floats
    float* partial = (float*)(ws + 32 * 1024); // NCHUNK * 4096 floats (256 KB)

    sinkhorn_init_c<<<N / 256, 256, 0, stream>>>(C);

    for (int t = 0; t < SINKHORN_ITERS; ++t) {
        sinkhorn_row_lse<<<N, 256, 0, stream>>>(M, C, R);
        sinkhorn_col_partial<<<dim3(N / 64, NCHUNK), 128, 0, stream>>>(M, R,
                                                                       partial);
        sinkhorn_col_finalize<<<N / 256, 256, 0, stream>>>(partial, C);
    }

    sinkhorn_final<<<(N * (size_t)N / 4) / 256, 256, 0, stream>>>(M, R, C, out);
}